// LongPatchBer_53841710022881
// MI455X (gfx1250) — compile-verified
//
#include <hip/hip_runtime.h>

// ---------------------------------------------------------------------------
// LongPatchBer on MI455X (gfx1250, wave32).
//
// out[p] = motif[p] * (w (*) profile)[p],  w[i] = sum_{c: idx_c == i} s_c,
// s_c = center_probs[c] / sum_k profile[k]*motif[idx_c + k - P]
//
// Pass 1: zero padded w (padding makes the conv staging branch-free).
// Pass 2: per-center gather (31 loads) + 1 float atomic into w.
// Pass 3: dense K=31 correlation via V_WMMA_F32_16X16X4_F32 (f32, exact),
//         fused with the motif multiply. One wave per 256 outputs; tile id
//         is readfirstlane'd so all guards are scalar branches (full EXEC
//         guaranteed at every WMMA).
// ---------------------------------------------------------------------------

#define KW 31
#define PW 15
#define TILE 256           // outputs per wave (16x16 WMMA tile)
#define WAVES_PER_BLOCK 8
#define STAGE 288          // w values staged per tile: [-15, +272]

typedef __attribute__((ext_vector_type(2))) float v2f;
typedef __attribute__((ext_vector_type(8))) float v8f;

__global__ void lpber_zero(float* __restrict__ p, int n) {
    int i = blockIdx.x * blockDim.x + threadIdx.x;
    int stride = gridDim.x * blockDim.x;
    for (; i < n; i += stride) p[i] = 0.0f;
}

// Pass 2: per-center normalization scalar -> atomic scatter into w.
__global__ __launch_bounds__(256) void lpber_centers(
        const float* __restrict__ prof, const float* __restrict__ mp,
        const float* __restrict__ cp, const int* __restrict__ idx,
        float* __restrict__ w, int C, int L) {
    int i = blockIdx.x * blockDim.x + threadIdx.x;
    if (i >= C) return;
    int c = idx[i];
    float s = 0.0f;
    if (c >= PW && c + PW < L) {
        // Interior center (the overwhelmingly common case): no clipping.
        const float* base = mp + (c - PW);
#pragma unroll
        for (int k = 0; k < KW; ++k)          // prof[k] uniform -> s_load
            s = fmaf(prof[k], base[k], s);
    } else {
#pragma unroll
        for (int k = 0; k < KW; ++k) {
            int j = c + k - PW;
            float m = (j >= 0 && j < L) ? mp[j] : 0.0f;
            s = fmaf(prof[k], m, s);
        }
    }
    atomicAdd(&w[c], cp[i] / s);
}

// Pass 3: 16x16 f32 WMMA tile = 256 consecutive outputs.
//   A[m,kk] = w[base + 16m - 15 + kk]   (16x48 Hankel, from LDS stage)
//   B[kk,n] = profile[kk - n]           (banded, zero outside [0,31))
//   out[base + 16m + n] = motif[...] * D[m,n]
// `w` points STAGE floats into a zeroed padded buffer, so staging never
// needs bounds checks.
__global__ __launch_bounds__(256) void lpber_conv_wmma(
        const float* __restrict__ prof, const float* __restrict__ mp,
        const float* __restrict__ w, float* __restrict__ out,
        int L, int tiles) {
    __shared__ float ldsW[WAVES_PER_BLOCK * STAGE];
    __shared__ float ldsProf[32];

    const int lane = threadIdx.x & 31;
    // threadIdx.x>>5 is wave-uniform; readfirstlane makes it an SGPR so all
    // tile-level guards compile to scalar branches (no exec masking).
    const int tile = __builtin_amdgcn_readfirstlane(
        blockIdx.x * WAVES_PER_BLOCK + (threadIdx.x >> 5));
    float* myW = &ldsW[(threadIdx.x >> 5) * STAGE];

    if (tile < tiles) {
        const float* src = w + (tile * TILE - PW);   // padded: always in-bounds
#pragma unroll
        for (int t = 0; t < STAGE / 32; ++t) {       // 9 loads, 9 ds_stores
            int li = lane + 32 * t;                  // 0..287
            myW[li] = src[li];
        }
    }
    if (threadIdx.x < KW) ldsProf[threadIdx.x] = prof[threadIdx.x];
    __syncthreads();
    if (tile >= tiles) return;                       // scalar branch

    const int half = lane >> 4;               // 0: lanes 0-15, 1: lanes 16-31
    const int mn = lane & 15;                 // M for A-frag, N for B-frag
    v8f acc = {};

#pragma unroll
    for (int q = 0; q < 12; ++q) {            // K = 48 = 12 x 4
        // ISA 32-bit A 16x4 layout: K = vgpr + 2*half, M = lane&15.
        int kk0 = 4 * q + 2 * half;
        v2f a, b;
        a.x = myW[16 * mn + kk0];             // A[mn, kk0]
        a.y = myW[16 * mn + kk0 + 1];         // A[mn, kk0+1]  (max idx 287)
        int d0 = kk0 - mn;                    // B[kk,n] = prof[kk-n]
        int d1 = d0 + 1;
        b.x = (d0 >= 0 && d0 < KW) ? ldsProf[d0] : 0.0f;
        b.y = (d1 >= 0 && d1 < KW) ? ldsProf[d1] : 0.0f;
        acc = __builtin_amdgcn_wmma_f32_16x16x4_f32(
                false, a, false, b, (short)0, acc, false, false);
    }

    const int base = tile * TILE;
    if (base + TILE <= L) {                   // scalar; taken for all tiles
        // when L % 256 == 0
#pragma unroll
        for (int v = 0; v < 8; ++v) {         // C/D layout: M = v + 8*half
            int j = base + 16 * (v + 8 * half) + mn;
            out[j] = acc[v] * mp[j];
        }
    } else {
#pragma unroll
        for (int v = 0; v < 8; ++v) {
            int j = base + 16 * (v + 8 * half) + mn;
            if (j < L) out[j] = acc[v] * mp[j];
        }
    }
}

// Fallback (ws too small for padded w): direct atomic scatter, reference math.
__global__ __launch_bounds__(256) void lpber_scatter_naive(
        const float* __restrict__ prof, const float* __restrict__ mp,
        const float* __restrict__ cp, const int* __restrict__ idx,
        float* __restrict__ out, int C, int L) {
    int i = blockIdx.x * blockDim.x + threadIdx.x;
    if (i >= C) return;
    int c = idx[i];
    float win[KW];
    float s = 0.0f;
#pragma unroll
    for (int k = 0; k < KW; ++k) {
        int j = c + k - PW;
        win[k] = (j >= 0 && j < L) ? mp[j] : 0.0f;
        s = fmaf(prof[k], win[k], s);
    }
    float val = cp[i] / s;
#pragma unroll
    for (int k = 0; k < KW; ++k) {
        int j = c + k - PW;
        if (j >= 0 && j < L) atomicAdd(&out[j], prof[k] * win[k] * val);
    }
}

extern "C" void kernel_launch(void* const* d_in, const int* in_sizes, int n_in,
                              void* d_out, int out_size, void* d_ws, size_t ws_size,
                              hipStream_t stream) {
    const float* prof = (const float*)d_in[0];   // [31]
    const float* mp   = (const float*)d_in[1];   // [L]
    const float* cp   = (const float*)d_in[2];   // [C]
    const int*   idx  = (const int*)d_in[3];     // [C]
    float* out = (float*)d_out;
    const int L = in_sizes[1];
    const int C = in_sizes[2];
    const int NPAD = L + 2 * STAGE;              // padded w size (floats)

    if (ws_size >= (size_t)NPAD * sizeof(float)) {
        float* wbuf = (float*)d_ws;
        float* w = wbuf + STAGE;                 // w[-STAGE .. L+STAGE) valid
        lpber_zero<<<4096, 256, 0, stream>>>(wbuf, NPAD);
        lpber_centers<<<(C + 255) / 256, 256, 0, stream>>>(prof, mp, cp, idx, w, C, L);
        int tiles = (L + TILE - 1) / TILE;
        int blocks = (tiles + WAVES_PER_BLOCK - 1) / WAVES_PER_BLOCK;
        lpber_conv_wmma<<<blocks, 256, 0, stream>>>(prof, mp, w, out, L, tiles);
    } else {
        lpber_zero<<<4096, 256, 0, stream>>>(out, L);
        lpber_scatter_naive<<<(C + 255) / 256, 256, 0, stream>>>(prof, mp, cp, idx, out, C, L);
    }
}